// SelfAttention_77919296684618
// MI455X (gfx1250) — compile-verified
//
#include <hip/hip_runtime.h>
#include <hip/hip_bf16.h>

// ---------------------------------------------------------------------------
// Sliding-window self-attention for MI455X (gfx1250), fp32 end to end.
// All matrix math on v_wmma_f32_16x16x4_f32 (wave32, 16x16 tiles).
// Register-blocked 2x2 per wave so each fragment load feeds 2 WMMAs.
// ---------------------------------------------------------------------------

typedef __attribute__((ext_vector_type(2))) float v2f;
typedef __attribute__((ext_vector_type(8))) float v8f;

#define HDIM   512
#define SEQ    2048
#define BATCH  2
#define WIN    128
#define PAD    64
#define MROWS  (BATCH * SEQ)          // 4096
#define JW     144                    // 128 window + 16 band slack

// WMMA fragment layouts (CDNA5 ISA §7.12.2):
//  A 16x4 (MxK):  lane&15 = M ; K = 2*(lane>>4) + vgpr  -> one float2 load
//  B 4x16 (KxN):  lane&15 = N ; K = 2*(lane>>4) + vgpr  -> one float2 load
//  C/D 16x16:     lane&15 = N ; M = vgpr + 8*(lane>>4)

static __device__ __forceinline__ v8f wmma4(v2f a, v2f b, v8f c) {
    return __builtin_amdgcn_wmma_f32_16x16x4_f32(
        false, a, false, b, (short)0, c, false, false);
}

// ---------------------------------------------------------------------------
// Kernel 1/4: Y[M,N] = X[M,K] @ W[N,K]^T + bias[N]      (M=4096, N=K=512)
// Block = 128 threads (4 waves). Block tile: 32 rows x 128 cols.
// Wave tile: 32x32 (2 M-tiles x 2 N-tiles) -> 1 fragment load per WMMA.
// ---------------------------------------------------------------------------
__global__ __launch_bounds__(128) void gemm16_bias(
    const float* __restrict__ X, const float* __restrict__ W,
    const float* __restrict__ bias, float* __restrict__ Y)
{
    __shared__ float ldsX[32 * HDIM];            // 64 KB

    const int tid    = threadIdx.x;
    const int wave   = tid >> 5;                 // 0..3
    const int lane   = tid & 31;
    const int laneLo = lane & 15;
    const int laneHi = lane >> 4;                // 0 or 1
    const int m0     = blockIdx.x * 32;
    const int nw     = blockIdx.y * 128 + wave * 32;

    // Stage contiguous 32xK slab of X into LDS (float4 vectorized).
    {
        const float4* src = (const float4*)(X + (size_t)m0 * HDIM);
        float4*       dst = (float4*)ldsX;
        #pragma unroll 4
        for (int i = tid; i < (32 * HDIM) / 4; i += 128) dst[i] = src[i];
    }
    __syncthreads();

    const int    na  = nw + laneLo;              // N-tile 0 column
    const int    nb  = na + 16;                  // N-tile 1 column
    const float* Wra = W + (size_t)na * HDIM;    // W is [N,K] row-major
    const float* Wrb = W + (size_t)nb * HDIM;

    v8f acc00 = {}, acc01 = {}, acc10 = {}, acc11 = {};

    for (int k0 = 0; k0 < HDIM; k0 += 64) {      // 256B-chunked K loop
        __builtin_prefetch(Wra + k0 + 64, 0, 1); // one prefetch per chunk
        __builtin_prefetch(Wrb + k0 + 64, 0, 1);
        #pragma unroll
        for (int kk = 0; kk < 64; kk += 4) {
            const int k = k0 + kk + 2 * laneHi;
            v2f a0 = *(const v2f*)&ldsX[(laneLo     ) * HDIM + k];
            v2f a1 = *(const v2f*)&ldsX[(laneLo + 16) * HDIM + k];
            v2f b0 = *(const v2f*)&Wra[k];
            v2f b1 = *(const v2f*)&Wrb[k];
            acc00 = wmma4(a0, b0, acc00);
            acc01 = wmma4(a0, b1, acc01);
            acc10 = wmma4(a1, b0, acc10);
            acc11 = wmma4(a1, b1, acc11);
        }
    }

    const float bna = bias[na];
    const float bnb = bias[nb];
    float* Y0 = Y + (size_t)(m0      + 8 * laneHi) * HDIM;
    float* Y1 = Y + (size_t)(m0 + 16 + 8 * laneHi) * HDIM;
    #pragma unroll
    for (int r = 0; r < 8; ++r) {
        Y0[(size_t)r * HDIM + na] = acc00[r] + bna;
        Y0[(size_t)r * HDIM + nb] = acc01[r] + bnb;
        Y1[(size_t)r * HDIM + na] = acc10[r] + bna;
        Y1[(size_t)r * HDIM + nb] = acc11[r] + bnb;
    }
}

// ---------------------------------------------------------------------------
// Kernel 2: RoPE exactly as the reference's interleaved formulation.
//   i in [0,256):  S_i = i<128 ? sin(t*f[2i])   : cos(t*f[2i-256])
//                  C_i = i<128 ? sin(t*f[2i+1]) : cos(t*f[2i-255])
//   out[i]     = x[2i]*C_i - x[2i+1]*S_i
//   out[i+256] = x[2i]*S_i + x[2i+1]*C_i
// with f[k] = 10000^(-k/256).
// ---------------------------------------------------------------------------
__global__ __launch_bounds__(256) void rope_kernel(
    const float* __restrict__ Qin, float* __restrict__ Qout)
{
    const int row = blockIdx.x;                  // b*S + t
    const int t   = row & (SEQ - 1);
    const int i   = threadIdx.x;                 // 0..255
    const float* q = Qin  + (size_t)row * HDIM;
    float*       o = Qout + (size_t)row * HDIM;

    const float x1 = q[2 * i];
    const float x2 = q[2 * i + 1];
    const float tt = (float)t;
    const float c  = 9.210340371976184f / 256.0f;   // ln(10000)/256

    float Sv, Cv;
    if (i < 128) {
        Sv = sinf(tt * expf(-(float)(2 * i)     * c));
        Cv = sinf(tt * expf(-(float)(2 * i + 1) * c));
    } else {
        Sv = cosf(tt * expf(-(float)(2 * i - 256) * c));
        Cv = cosf(tt * expf(-(float)(2 * i - 255) * c));
    }
    o[i]       = x1 * Cv - x2 * Sv;
    o[i + 256] = x1 * Sv + x2 * Cv;
}

// ---------------------------------------------------------------------------
// Kernel 3: sliding-window attention for one 16-query block.
// Band trick: for queries t0..t0+15, key positions span p0..p0+142 where
// p0 = t0-64. Score band S'[q,j] = rQ[t0+q] . rK[p0+j]; row q's window is
// j in [q, q+127]. OOB keys give score 0 (reference zero-pads K), and those
// zeros DO participate in softmax; OOB V rows are zero.
// ---------------------------------------------------------------------------
__global__ __launch_bounds__(128) void sw_attention(
    const float* __restrict__ rQ, const float* __restrict__ rK,
    const float* __restrict__ V,  float* __restrict__ Out)
{
    __shared__ float ldsQ[16 * HDIM];            // 32 KB
    __shared__ float ldsS[16 * JW];              // 9 KB scores
    __shared__ float ldsW[16 * JW];              // 9 KB softmax weights
    __shared__ float red[16 * 8];                // reduction scratch

    const int tid    = threadIdx.x;
    const int wave   = tid >> 5;
    const int lane   = tid & 31;
    const int laneLo = lane & 15;
    const int laneHi = lane >> 4;

    const int b  = blockIdx.x / (SEQ / 16);
    const int t0 = (blockIdx.x % (SEQ / 16)) * 16;
    const int p0 = t0 - PAD;

    const float* Qb = rQ + (size_t)b * SEQ * HDIM;
    const float* Kb = rK + (size_t)b * SEQ * HDIM;
    const float* Vb = V  + (size_t)b * SEQ * HDIM;

    // Stage rQ tile, zero weight band.
    {
        const float4* src = (const float4*)(Qb + (size_t)t0 * HDIM);
        float4*       dst = (float4*)ldsQ;
        #pragma unroll 4
        for (int i = tid; i < (16 * HDIM) / 4; i += 128) dst[i] = src[i];
        for (int i = tid; i < 16 * JW; i += 128) ldsW[i] = 0.0f;
    }
    __syncthreads();

    const float scale = 0.044194173824159216f;   // 1/sqrt(512)

    // ---- Phase 1: score band via WMMA. Waves take j-tile pairs (w, w+4)
    //      sharing one A fragment per k-step; wave 0 also does tile 8. ----
    {
        const int jA = wave * 16 + laneLo;             // tile jt = wave
        const int jB = jA + 64;                        // tile jt = wave+4
        const int pA = p0 + jA,  pB = p0 + jB;
        const int cA = pA < 0 ? 0 : (pA >= SEQ ? SEQ - 1 : pA);
        const int cB = pB < 0 ? 0 : (pB >= SEQ ? SEQ - 1 : pB);
        const float mA = (pA >= 0 && pA < SEQ) ? 1.0f : 0.0f;
        const float mB = (pB >= 0 && pB < SEQ) ? 1.0f : 0.0f;
        const float* KrA = Kb + (size_t)cA * HDIM;
        const float* KrB = Kb + (size_t)cB * HDIM;

        v8f accA = {}, accB = {};
        for (int k0 = 0; k0 < HDIM; k0 += 4) {
            const int k = k0 + 2 * laneHi;
            v2f a  = *(const v2f*)&ldsQ[laneLo * HDIM + k];
            v2f kA = *(const v2f*)&KrA[k] * mA;
            v2f kB = *(const v2f*)&KrB[k] * mB;
            accA = wmma4(a, kA, accA);
            accB = wmma4(a, kB, accB);
        }
        #pragma unroll
        for (int r = 0; r < 8; ++r) {
            ldsS[(r + 8 * laneHi) * JW + (wave     ) * 16 + laneLo] = accA[r] * scale;
            ldsS[(r + 8 * laneHi) * JW + (wave + 4 ) * 16 + laneLo] = accB[r] * scale;
        }
    }
    if (wave == 0) {                                   // j-tile 8 (wave-uniform)
        const int j  = 128 + laneLo;
        const int p  = p0 + j;
        const int pc = p < 0 ? 0 : (p >= SEQ ? SEQ - 1 : p);
        const float m = (p >= 0 && p < SEQ) ? 1.0f : 0.0f;
        const float* Kr = Kb + (size_t)pc * HDIM;
        v8f acc = {};
        for (int k0 = 0; k0 < HDIM; k0 += 4) {
            const int k = k0 + 2 * laneHi;
            v2f a  = *(const v2f*)&ldsQ[laneLo * HDIM + k];
            v2f kb = *(const v2f*)&Kr[k] * m;
            acc = wmma4(a, kb, acc);
        }
        #pragma unroll
        for (int r = 0; r < 8; ++r)
            ldsS[(r + 8 * laneHi) * JW + 128 + laneLo] = acc[r] * scale;
    }
    __syncthreads();

    // ---- Phase 2: softmax over each row's 128-wide window (8 thr/row) ----
    {
        const int row   = tid >> 3;              // 0..15
        const int seg   = tid & 7;               // 0..7
        const int jbase = row + seg * 16;        // window offsets seg*16..+15

        float sv[16];
        float mx = -3.4e38f;
        #pragma unroll
        for (int ii = 0; ii < 16; ++ii) {
            sv[ii] = ldsS[row * JW + jbase + ii];
            mx = fmaxf(mx, sv[ii]);
        }
        red[row * 8 + seg] = mx;
        __syncthreads();
        float rmx = red[row * 8];
        #pragma unroll
        for (int u = 1; u < 8; ++u) rmx = fmaxf(rmx, red[row * 8 + u]);
        __syncthreads();

        float lsum = 0.0f;
        #pragma unroll
        for (int ii = 0; ii < 16; ++ii) {
            sv[ii] = __expf(sv[ii] - rmx);
            lsum  += sv[ii];
        }
        red[row * 8 + seg] = lsum;
        __syncthreads();
        float rsum = 0.0f;
        #pragma unroll
        for (int u = 0; u < 8; ++u) rsum += red[row * 8 + u];
        const float inv = 1.0f / rsum;
        #pragma unroll
        for (int ii = 0; ii < 16; ++ii)
            ldsW[row * JW + jbase + ii] = sv[ii] * inv;
    }
    __syncthreads();

    // ---- Phase 3: Out[16x512] = Wgt[16x144] x V[144x512] via WMMA.
    //      N-tiles in pairs so one A fragment feeds 2 WMMAs per j-step. ----
    for (int nt = wave * 2; nt < HDIM / 16; nt += 8) {
        const int na = nt * 16 + laneLo;
        const int nb = na + 16;
        v8f accA = {}, accB = {};
        for (int j0 = 0; j0 < JW; j0 += 4) {
            v2f a = *(const v2f*)&ldsW[laneLo * JW + j0 + 2 * laneHi];
            const int pr0 = p0 + j0 + 2 * laneHi;
            const int pr1 = pr0 + 1;
            const int c0  = pr0 < 0 ? 0 : (pr0 >= SEQ ? SEQ - 1 : pr0);
            const int c1  = pr1 < 0 ? 0 : (pr1 >= SEQ ? SEQ - 1 : pr1);
            const float m0f = (pr0 >= 0 && pr0 < SEQ) ? 1.0f : 0.0f;
            const float m1f = (pr1 >= 0 && pr1 < SEQ) ? 1.0f : 0.0f;
            const float* V0 = Vb + (size_t)c0 * HDIM;
            const float* V1 = Vb + (size_t)c1 * HDIM;
            v2f bA, bB;
            bA.x = V0[na] * m0f;  bA.y = V1[na] * m1f;
            bB.x = V0[nb] * m0f;  bB.y = V1[nb] * m1f;
            accA = wmma4(a, bA, accA);
            accB = wmma4(a, bB, accB);
        }
        float* Op = Out + (size_t)b * SEQ * HDIM + (size_t)(t0 + 8 * laneHi) * HDIM;
        #pragma unroll
        for (int r = 0; r < 8; ++r) {
            Op[(size_t)r * HDIM + na] = accA[r];
            Op[(size_t)r * HDIM + nb] = accB[r];
        }
    }
}

// ---------------------------------------------------------------------------
extern "C" void kernel_launch(void* const* d_in, const int* in_sizes, int n_in,
                              void* d_out, int out_size, void* d_ws, size_t ws_size,
                              hipStream_t stream)
{
    const float* x  = (const float*)d_in[0];
    const float* Wq = (const float*)d_in[1];
    const float* bq = (const float*)d_in[2];
    const float* Wk = (const float*)d_in[3];
    const float* bk = (const float*)d_in[4];
    const float* Wv = (const float*)d_in[5];
    const float* bv = (const float*)d_in[6];
    const float* Wo = (const float*)d_in[7];
    const float* bo = (const float*)d_in[8];

    const size_t MAT = (size_t)MROWS * HDIM;     // 2M floats per matrix
    float* Qm   = (float*)d_ws;
    float* Km   = Qm  + MAT;
    float* Vm   = Km  + MAT;
    float* rQm  = Vm  + MAT;
    float* rKm  = rQm + MAT;
    float* attn = rKm + MAT;                     // 48 MB total workspace

    const dim3 gemmGrid(MROWS / 32, HDIM / 128); // 128 x 4 blocks
    gemm16_bias<<<gemmGrid, 128, 0, stream>>>(x, Wq, bq, Qm);
    gemm16_bias<<<gemmGrid, 128, 0, stream>>>(x, Wk, bk, Km);
    gemm16_bias<<<gemmGrid, 128, 0, stream>>>(x, Wv, bv, Vm);

    rope_kernel<<<MROWS, 256, 0, stream>>>(Qm, rQm);
    rope_kernel<<<MROWS, 256, 0, stream>>>(Km, rKm);

    sw_attention<<<BATCH * (SEQ / 16), 128, 0, stream>>>(rQm, rKm, Vm, attn);

    gemm16_bias<<<gemmGrid, 128, 0, stream>>>(attn, Wo, bo, (float*)d_out);
}